// GTNs_61692910240527
// MI455X (gfx1250) — compile-verified
//
#include <hip/hip_runtime.h>
#include <cstdint>
#include <cstddef>

// ---------------------------------------------------------------------------
// GTN forward (gather-scale-scatter-add), MI455X / gfx1250.
// Memory-bound op: H (51.2MB) + out (51.2MB) are L2-resident; edges stream.
// Strategy: async-to-LDS double-buffered edge staging (ASYNCcnt pipeline),
// wave-cooperative 256B-coalesced gathers/scatters, hw f32 global atomics
// (RMW resolved at L2), global_prefetch lookahead. WMMA intentionally unused:
// no dense matmul exists here (0.64 GFLOP vs ~2.6GB of L2 traffic).
//
// Round-1 delta: E (500000) is a multiple of the 32-edge chunk, so the edge
// type t — and therefore the softmax scale pick — is uniform per chunk.
// Hoist the divide + two cndmask chains out of the inner loop (~10 VALU/edge).
// ---------------------------------------------------------------------------

#define kN          100000
#define kF          64
#define kT          5
#define kE          500000
#define kTotalEdges (uint32_t)(kT * kE)        // 2,500,000 (multiple of 32)
#define kNF         (uint32_t)(kN * kF)        // 6,400,000
#define kCNF        (uint32_t)(2u * kNF)       // 12,800,000
#define kChunk      32
#define kNumChunks  (kTotalEdges / kChunk)     // 78,125
#define kWPB        8                          // waves per block (256 threads)

typedef __attribute__((address_space(3))) unsigned char LdsByte;

static __device__ __forceinline__ uint32_t lds_off32(void* p) {
  // generic -> LDS(3) addrspacecast; AS(3) pointers are 32-bit LDS offsets
  return (uint32_t)(uintptr_t)(LdsByte*)p;
}

// GLOBAL_LOAD_ASYNC_TO_LDS_B32, GVS mode: each lane copies the dword at saddr + vaddr to LDS offset vdst
static __device__ __forceinline__ void async_g2l_b32(uint32_t lds, uint32_t voff,
                                                     uint64_t sbase) {
  asm volatile("global_load_async_to_lds_b32 %0, %1, %2"
               :
               : "v"(lds), "v"(voff), "s"(sbase)
               : "memory");
}

static __device__ __forceinline__ float pick5(uint32_t t, const float* f) {
  float r = f[0];
  r = (t == 1u) ? f[1] : r;
  r = (t == 2u) ? f[2] : r;
  r = (t == 3u) ? f[3] : r;
  r = (t == 4u) ? f[4] : r;
  return r;
}

static __device__ __forceinline__ void softmax5(const float* w, float* f) {
  float m = fmaxf(fmaxf(fmaxf(w[0], w[1]), fmaxf(w[2], w[3])), w[4]);
  float s = 0.f;
#pragma unroll
  for (int t = 0; t < 5; ++t) { f[t] = __expf(w[t] - m); s += f[t]; }
  float r = 1.0f / s;
#pragma unroll
  for (int t = 0; t < 5; ++t) f[t] *= r;
}

// ---------------------------------------------------------------------------
// Kernel 1: zero the accumulator region of d_out, write filt tail (10 floats)
// ---------------------------------------------------------------------------
__global__ __launch_bounds__(256) void gtn_zero_filt(float* __restrict__ out,
                                                     const float* __restrict__ weight) {
  uint32_t i4 = 4u * (blockIdx.x * blockDim.x + threadIdx.x);
  if (i4 < kCNF) {
    float4 z = make_float4(0.f, 0.f, 0.f, 0.f);
    *(float4*)(out + i4) = z;
  }
  if (blockIdx.x == 0 && threadIdx.x == 0) {
    float wr[10];
#pragma unroll
    for (int i = 0; i < 10; ++i) wr[i] = weight[i];
    float f0[5], f1[5];
    softmax5(wr, f0);
    softmax5(wr + 5, f1);
#pragma unroll
    for (int t = 0; t < 5; ++t) {
      out[kCNF + t]     = f0[t];
      out[kCNF + 5 + t] = f1[t];
    }
  }
}

// ---------------------------------------------------------------------------
// Kernel 2: edge scatter. One wave processes 32 staged edges at a time;
// per edge, lanes 0..31 each own a float2 slice of the 64-wide feature row.
// ---------------------------------------------------------------------------
__global__ __launch_bounds__(256) void gtn_scatter(const float* __restrict__ H,
                                                   const int* __restrict__ eidx,
                                                   const float* __restrict__ evals,
                                                   const float* __restrict__ weight,
                                                   float* __restrict__ out) {
  __shared__ int   sh_row[kWPB][2][kChunk];
  __shared__ int   sh_col[kWPB][2][kChunk];
  __shared__ float sh_val[kWPB][2][kChunk];

  const uint32_t lane    = threadIdx.x & 31u;
  const uint32_t w       = threadIdx.x >> 5;
  const uint32_t waveGid = blockIdx.x * (blockDim.x >> 5) + w;
  const uint32_t nWaves  = gridDim.x * (blockDim.x >> 5);

  // filt in registers (uniform scalar loads + cheap VALU softmax)
  float wr[10];
#pragma unroll
  for (int i = 0; i < 10; ++i) wr[i] = weight[i];
  float f0[5], f1[5];
  softmax5(wr, f0);
  softmax5(wr + 5, f1);

  const uint64_t idxBase = (uint64_t)(uintptr_t)eidx;
  const uint64_t valBase = (uint64_t)(uintptr_t)evals;
  const float* __restrict__ H0 = H;
  const float* __restrict__ H1 = H + kNF;
  float* __restrict__ O0 = out;
  float* __restrict__ O1 = out + kNF;
  const uint32_t fo = 2u * lane;  // this lane's float2 slot in the 64-wide row

  // Stage 32 edges (row, col, val) for this wave into LDS buffer b.
  auto stage = [&](uint32_t chunk, int b) {
    uint32_t g  = chunk * (uint32_t)kChunk + lane;      // < kTotalEdges always
    uint32_t t  = g / (uint32_t)kE;
    uint32_t e  = g - t * (uint32_t)kE;
    uint32_t ro = 4u * (t * (uint32_t)(2 * kE) + e);    // edge_index[t][0][e]
    async_g2l_b32(lds_off32(&sh_row[w][b][lane]), ro, idxBase);
    async_g2l_b32(lds_off32(&sh_col[w][b][lane]), ro + 4u * (uint32_t)kE, idxBase);
    async_g2l_b32(lds_off32(&sh_val[w][b][lane]), 4u * g, valBase);
  };

  auto process = [&](uint32_t chunk, int b) {
    // kE % kChunk == 0  =>  a chunk never crosses an edge-type boundary:
    // t (and therefore the softmax scale) is uniform for the whole chunk.
    const uint32_t t  = (chunk * (uint32_t)kChunk) / (uint32_t)kE;
    const float    s0 = pick5(t, f0);
    const float    s1 = pick5(t, f1);
#pragma unroll 4
    for (int i = 0; i < kChunk; ++i) {
      // broadcast reads (uniform LDS address; compiler packs into ds_load_b128)
      const int   row = sh_row[w][b][i];
      const int   col = sh_col[w][b][i];
      const float val = sh_val[w][b][i];
      // lookahead prefetch of a future edge's H rows (cols already in LDS)
      const int pcol = sh_col[w][b][(i + 8) & (kChunk - 1)];
      __builtin_prefetch(H0 + (uint32_t)pcol * (uint32_t)kF + fo, 0, 1);
      __builtin_prefetch(H1 + (uint32_t)pcol * (uint32_t)kF + fo, 0, 1);

      const float a0 = val * s0;
      const float a1 = val * s1;

      const float2 h0 = *(const float2*)(H0 + (uint32_t)col * (uint32_t)kF + fo);
      const float2 h1 = *(const float2*)(H1 + (uint32_t)col * (uint32_t)kF + fo);

      float* p0 = O0 + (uint32_t)row * (uint32_t)kF + fo;
      float* p1 = O1 + (uint32_t)row * (uint32_t)kF + fo;
      unsafeAtomicAdd(p0 + 0, a0 * h0.x);   // hw global_atomic_add_f32
      unsafeAtomicAdd(p0 + 1, a0 * h0.y);
      unsafeAtomicAdd(p1 + 0, a1 * h1.x);
      unsafeAtomicAdd(p1 + 1, a1 * h1.y);
    }
  };

  // Double-buffered ASYNCcnt pipeline: 3 async loads per staged chunk.
  uint32_t chunk = waveGid;
  int buf = 0;
  if (chunk < kNumChunks) stage(chunk, 0);
  while (chunk < kNumChunks) {
    const uint32_t next = chunk + nWaves;
    if (next < kNumChunks) {
      stage(next, buf ^ 1);
      // up to 6 outstanding; in-order completion => waiting to <=3 guarantees
      // the current chunk's 3 async loads have landed in LDS
      asm volatile("s_wait_asynccnt 0x3" ::: "memory");
    } else {
      asm volatile("s_wait_asynccnt 0x0" ::: "memory");
    }
    process(chunk, buf);
    chunk = next;
    buf ^= 1;
  }
}

// ---------------------------------------------------------------------------
extern "C" void kernel_launch(void* const* d_in, const int* in_sizes, int n_in,
                              void* d_out, int out_size, void* d_ws, size_t ws_size,
                              hipStream_t stream) {
  const float* H      = (const float*)d_in[0];  // [2, 100000, 64]
  const int*   eidx   = (const int*)d_in[1];    // [5, 2, 500000]
  const float* evals  = (const float*)d_in[2];  // [5, 500000]
  const float* weight = (const float*)d_in[3];  // [2, 5]
  float* out = (float*)d_out;                   // 12,800,000 Hs + 10 filt

  (void)in_sizes; (void)n_in; (void)out_size; (void)d_ws; (void)ws_size;

  // 1) zero accumulators + write filt tail
  const uint32_t zblocks = (kCNF / 4u + 255u) / 256u;  // 12,500
  gtn_zero_filt<<<dim3(zblocks), dim3(256), 0, stream>>>(out, weight);

  // 2) scatter-add over all edges
  gtn_scatter<<<dim3(1024), dim3(256), 0, stream>>>(H, eidx, evals, weight, out);
}